// GCN_55224689492101
// MI455X (gfx1250) — compile-verified
//
#include <hip/hip_runtime.h>
#include <hip/hip_bf16.h>

#define NFEAT 128
#define NCLASS 40

typedef float v2f __attribute__((ext_vector_type(2)));
typedef float v8f __attribute__((ext_vector_type(8)));

// ---------- wave32 reductions ----------
static __device__ __forceinline__ float wave_sum(float v) {
#pragma unroll
  for (int m = 16; m >= 1; m >>= 1) v += __shfl_xor(v, m, 32);
  return v;
}
static __device__ __forceinline__ float wave_max(float v) {
#pragma unroll
  for (int m = 16; m >= 1; m >>= 1) v = fmaxf(v, __shfl_xor(v, m, 32));
  return v;
}

// ---------- LayerNorm over 128-wide rows: one wave per row, 4 floats/lane ----------
__global__ __launch_bounds__(256) void ln128_kernel(
    const float* __restrict__ x, const float* __restrict__ g,
    const float* __restrict__ b, float* __restrict__ out, int nrows) {
  int row  = (int)((blockIdx.x * blockDim.x + threadIdx.x) >> 5);
  int lane = threadIdx.x & 31;
  if (row >= nrows) return;
  float4 v = ((const float4*)(x + (size_t)row * NFEAT))[lane];
  float mu = wave_sum(v.x + v.y + v.z + v.w) * (1.0f / NFEAT);
  float dx = v.x - mu, dy = v.y - mu, dz = v.z - mu, dw = v.w - mu;
  float var = wave_sum(dx * dx + dy * dy + dz * dz + dw * dw) * (1.0f / NFEAT);
  float rs = rsqrtf(var + 1e-5f);
  float4 gg = ((const float4*)g)[lane];
  float4 bb = ((const float4*)b)[lane];
  float4 o;
  o.x = dx * rs * gg.x + bb.x;
  o.y = dy * rs * gg.y + bb.y;
  o.z = dz * rs * gg.z + bb.z;
  o.w = dw * rs * gg.w + bb.w;
  ((float4*)(out + (size_t)row * NFEAT))[lane] = o;
}

// ---------- fused L2-normalize + LayerNorm + ReLU (128-wide rows) ----------
__global__ __launch_bounds__(256) void l2_ln_relu_kernel(
    const float* __restrict__ in, const float* __restrict__ g,
    const float* __restrict__ b, float* __restrict__ out, int nrows) {
  int row  = (int)((blockIdx.x * blockDim.x + threadIdx.x) >> 5);
  int lane = threadIdx.x & 31;
  if (row >= nrows) return;
  float4 v = ((const float4*)(in + (size_t)row * NFEAT))[lane];
  float ss = wave_sum(v.x * v.x + v.y * v.y + v.z * v.z + v.w * v.w);
  float scale = 1.0f / fmaxf(sqrtf(ss), 1e-12f);
  v.x *= scale; v.y *= scale; v.z *= scale; v.w *= scale;
  float mu = wave_sum(v.x + v.y + v.z + v.w) * (1.0f / NFEAT);
  float dx = v.x - mu, dy = v.y - mu, dz = v.z - mu, dw = v.w - mu;
  float var = wave_sum(dx * dx + dy * dy + dz * dz + dw * dw) * (1.0f / NFEAT);
  float rs = rsqrtf(var + 1e-5f);
  float4 gg = ((const float4*)g)[lane];
  float4 bb = ((const float4*)b)[lane];
  float4 o;
  o.x = fmaxf(dx * rs * gg.x + bb.x, 0.0f);
  o.y = fmaxf(dy * rs * gg.y + bb.y, 0.0f);
  o.z = fmaxf(dz * rs * gg.z + bb.z, 0.0f);
  o.w = fmaxf(dw * rs * gg.w + bb.w, 0.0f);
  ((float4*)(out + (size_t)row * NFEAT))[lane] = o;
}

// ---------- dual fp32 WMMA GEMM: outL = A*Wl^T + bias, outR = A*Wr^T ----------
// A: [M,128] row-major.  Wl/Wr: [NOUT,128] row-major (so B[k][n] = W[n][k]).
// Block = 256 threads (8 waves). Each block owns MT consecutive M-tiles:
//   * B fragments (16x128 per wave per weight) are loaded ONCE into registers
//     and amortized over all MT tiles (10x less L2 B-traffic).
//   * The 8 KB fp32 A tile is staged into LDS via the CDNA5 async path
//     (global_load_async_to_lds_b128, ASYNCcnt), DOUBLE-BUFFERED: tile t+1 is
//     prefetched while WMMAs consume tile t. s_wait_asynccnt 0x2 relies on the
//     ISA's in-order async completion: the 2 newest chunks (prefetch t+1) may
//     remain in flight while the previous tile is guaranteed landed.
//   * 132-float LDS row pitch keeps the 16-row fragment reads conflict-free.
// Fragment layouts per ISA 7.12.2 (f32):
//   A 16x4 : lane L -> row = L&15, K pair = 2*(L>>4)
//   B 4x16 : lane L -> col n = L&15, K rows 2*(L>>4)+{0,1}
//   C/D    : VGPR v -> row m0 + v + 8*(L>>4), col n = L&15
#define APITCH 132
#define KSTEPS (NFEAT / 4)

template <int NOUT, int NTILES, bool MASKED, int MT>
__global__ __launch_bounds__(256) void gemm_dual_wmma(
    const float* __restrict__ A, const float* __restrict__ Wl,
    const float* __restrict__ bias, const float* __restrict__ Wr,
    float* __restrict__ outL, float* __restrict__ outR, int totalTiles) {
  __shared__ float tileA[2][16 * APITCH];
  const int tid  = threadIdx.x;
  const int w    = tid >> 5;
  const int lane = tid & 31;
  const int base = blockIdx.x * MT;
  int tiles = totalTiles - base;
  if (tiles > MT) tiles = MT;

  const int r  = lane & 15;
  const int hi = lane >> 4;
  const int n  = (w << 4) + r;
  const int nc = MASKED ? ((n < NOUT) ? n : (NOUT - 1)) : n;
  const float nmask = (!MASKED || n < NOUT) ? 1.0f : 0.0f;

  // ---- register-resident B fragments for both weights, reused for MT tiles ----
  v2f bl[KSTEPS], br[KSTEPS];
  if (w < NTILES) {
    const float* blp = Wl + (size_t)nc * NFEAT + 2 * hi;
    const float* brp = Wr + (size_t)nc * NFEAT + 2 * hi;
#pragma unroll
    for (int ks = 0; ks < KSTEPS; ++ks) {
      bl[ks] = *(const v2f*)(blp + 4 * ks);
      br[ks] = *(const v2f*)(brp + 4 * ks);
      if (MASKED) { bl[ks] *= nmask; br[ks] *= nmask; }
    }
  }

  // ---- async stage of one A tile: 512 x 16B chunks, 2 per thread ----
  auto stage = [&](int bufIdx, int mtile) {
#pragma unroll
    for (int i = 0; i < 2; ++i) {
      int c    = tid + i * 256;
      int row  = c >> 5;
      int col4 = (c & 31) << 2;  // float index within the row, step 4
      unsigned lds = (unsigned)(uintptr_t)&tileA[bufIdx][row * APITCH + col4];
      const float* gp = A + ((size_t)mtile * 16 + row) * NFEAT + col4;
      asm volatile("global_load_async_to_lds_b128 %0, %1, off"
                   :: "v"(lds), "v"(gp)
                   : "memory");
    }
  };

  stage(0, base);
  for (int t = 0; t < tiles; ++t) {
    const int cur = t & 1;
    if (t + 1 < tiles) {
      stage(cur ^ 1, base + t + 1);                  // overlap next-tile DMA
      asm volatile("s_wait_asynccnt 0x2" ::: "memory");  // tile t landed
    } else {
      asm volatile("s_wait_asynccnt 0x0" ::: "memory");
    }
    __syncthreads();  // all waves' chunks of tileA[cur] visible

    if (w < NTILES) {  // wave-uniform; EXEC all-1s for WMMA waves
      const float* alp = &tileA[cur][r * APITCH + 2 * hi];
      v8f cl = {};
      v8f cr = {};
#pragma unroll
      for (int ks = 0; ks < KSTEPS; ++ks) {
        v2f a = *(const v2f*)(alp + 4 * ks);  // ds_load, conflict-free
        cl = __builtin_amdgcn_wmma_f32_16x16x4_f32(false, a, false, bl[ks],
                                                   (short)0, cl, false, false);
        cr = __builtin_amdgcn_wmma_f32_16x16x4_f32(false, a, false, br[ks],
                                                   (short)0, cr, false, false);
      }
      if (!MASKED || n < NOUT) {
        const int m0 = (base + t) << 4;
        float bv = bias[n];
#pragma unroll
        for (int v = 0; v < 8; ++v) {
          int m = m0 + v + 8 * hi;
          outL[(size_t)m * NOUT + n] = cl[v] + bv;
          outR[(size_t)m * NOUT + n] = cr[v];
        }
      }
    }
    __syncthreads();  // reads of tileA[cur] done before t+1 overwrites it
  }
}

// ---------- edge message scatter: one wave per edge, lane-strided features ----------
template <int F>
__global__ __launch_bounds__(256) void scatter_kernel(
    const long long* __restrict__ ei, const float* __restrict__ ew,
    const float* __restrict__ msg, float* __restrict__ acc, int E) {
  int e    = (int)((blockIdx.x * blockDim.x + threadIdx.x) >> 5);
  int lane = threadIdx.x & 31;
  if (e >= E) return;
  int src = (int)ei[e];
  int dst = (int)ei[(size_t)E + e];
  float wgt = ew[e];
  const float* in = msg + (size_t)src * F;
  float* out = acc + (size_t)dst * F;
#pragma unroll
  for (int f = lane; f < F; f += 32)
    unsafeAtomicAdd(out + f, wgt * in[f]);  // native global_atomic_add_f32
}

// ---------- fused L2-normalize + log_softmax over 40 classes ----------
__global__ __launch_bounds__(256) void l2_logsoftmax_kernel(
    const float* __restrict__ in, float* __restrict__ out, int nrows) {
  int row  = (int)((blockIdx.x * blockDim.x + threadIdx.x) >> 5);
  int lane = threadIdx.x & 31;
  if (row >= nrows) return;
  const float* r = in + (size_t)row * NCLASS;
  bool v1 = (lane + 32) < NCLASS;  // lanes 0..7 carry a 2nd element
  float a0 = r[lane];
  float a1 = v1 ? r[lane + 32] : 0.0f;
  float ss = wave_sum(a0 * a0 + (v1 ? a1 * a1 : 0.0f));
  float scale = 1.0f / fmaxf(sqrtf(ss), 1e-12f);
  a0 *= scale; a1 *= scale;
  float mx = wave_max(fmaxf(a0, v1 ? a1 : -3.402823e38f));
  float es = wave_sum(expf(a0 - mx) + (v1 ? expf(a1 - mx) : 0.0f));
  float lse = mx + logf(es);
  float* o = out + (size_t)row * NCLASS;
  o[lane] = a0 - lse;
  if (v1) o[lane + 32] = a1 - lse;
}

extern "C" void kernel_launch(void* const* d_in, const int* in_sizes, int n_in,
                              void* d_out, int out_size, void* d_ws, size_t ws_size,
                              hipStream_t stream) {
  const float*     x    = (const float*)d_in[0];
  const long long* ei   = (const long long*)d_in[1];  // [2,E] int64
  const float*     ew   = (const float*)d_in[2];
  const float*     ln0g = (const float*)d_in[3];
  const float*     ln0b = (const float*)d_in[4];
  const float*     W0l  = (const float*)d_in[5];
  const float*     b0l  = (const float*)d_in[6];
  const float*     W0r  = (const float*)d_in[7];
  const float*     ln1g = (const float*)d_in[8];
  const float*     ln1b = (const float*)d_in[9];
  const float*     W1l  = (const float*)d_in[10];
  const float*     b1l  = (const float*)d_in[11];
  const float*     W1r  = (const float*)d_in[12];
  float*           out  = (float*)d_out;

  const int N = in_sizes[0] / NFEAT;
  const int E = in_sizes[2];

  // Workspace: three N*128 fp32 buffers (ping-pong reuse keeps footprint ~61 MB).
  const size_t buf = (size_t)N * NFEAT;
  float* bufA = (float*)d_ws;                 // LN(x), later reused as h2
  float* bufB = bufA + buf;                   // lin_l projection, later proj2|acc2
  float* bufC = bufB + buf;                   // conv-1 accumulator (root + scatter)
  float* proj2 = bufB;                        // [N,40]
  float* acc2  = bufB + (size_t)N * NCLASS;   // [N,40]

  const int MT         = 10;            // M-tiles per GEMM block
  const int rowBlocks  = (N + 7) / 8;   // 8 rows (waves) per 256-thread block
  const int mTiles     = (N + 15) / 16; // 40000 -> 2500
  const int gemmBlocks = (mTiles + MT - 1) / MT;  // 250
  const int edgeBlocks = (E + 7) / 8;   // 8 edges (waves) per block

  // 1) h = LayerNorm(x)
  ln128_kernel<<<rowBlocks, 256, 0, stream>>>(x, ln0g, ln0b, bufA, N);
  // 2) proj = h@W0l^T + b0l (bufB); root = h@W0r^T (bufC)
  gemm_dual_wmma<NFEAT, 8, false, 10><<<gemmBlocks, 256, 0, stream>>>(
      bufA, W0l, b0l, W0r, bufB, bufC, mTiles);
  // 3) bufC += scatter_add(proj[src] * ew -> dst)
  scatter_kernel<NFEAT><<<edgeBlocks, 256, 0, stream>>>(ei, ew, bufB, bufC, E);
  // 4) h2 = relu(LN1(l2normalize(bufC)))
  l2_ln_relu_kernel<<<rowBlocks, 256, 0, stream>>>(bufC, ln1g, ln1b, bufA, N);
  // 5) proj2 = h2@W1l^T + b1l; acc2 = h2@W1r^T   (Nout=40 -> 3 padded 16-tiles)
  gemm_dual_wmma<NCLASS, 3, true, 10><<<gemmBlocks, 256, 0, stream>>>(
      bufA, W1l, b1l, W1r, proj2, acc2, mTiles);
  // 6) acc2 += scatter_add(proj2[src] * ew -> dst)
  scatter_kernel<NCLASS><<<edgeBlocks, 256, 0, stream>>>(ei, ew, proj2, acc2, E);
  // 7) out = log_softmax(l2normalize(acc2))
  l2_logsoftmax_kernel<<<rowBlocks, 256, 0, stream>>>(acc2, out, N);
}